// ContactMapGeneration_69715909148738
// MI455X (gfx1250) — compile-verified
//
#include <hip/hip_runtime.h>
#include <cstdint>

// ---------------------------------------------------------------------------
// Problem constants (reference: B=16, N=100000, npoint=1024)
// ---------------------------------------------------------------------------
#define BATCHES       16
#define NPTS          100000
#define NPOINT        1024
#define WGS_PER_BATCH 4                       // 4 WGPs cooperate per batch
#define NPPW          (NPTS / WGS_PER_BATCH)  // 25000 points per workgroup
#define THREADS       1024                    // 32 wave32 waves per WG
#define NPT           ((NPPW + THREADS - 1) / THREADS)  // 25 points/thread
#define BIGF          1.0e10f
#define SLICE_BYTES   (NPPW * 3 * sizeof(float))        // 300000 B of LDS

// ---------------------------------------------------------------------------
// CDNA5 async global->LDS copy (ASYNCcnt path). Probe-verified signature:
//   void __builtin_amdgcn_global_load_async_to_lds_b128(
//        v4i addrspace(1)* gsrc, v4i addrspace(3)* ldst, int off, int cpol)
// ---------------------------------------------------------------------------
typedef int v4i __attribute__((vector_size(4 * sizeof(int))));
typedef __attribute__((address_space(1))) v4i* gbl_v4i_ptr;
typedef __attribute__((address_space(3))) v4i* lds_v4i_ptr;

#if defined(__AMDGCN__) && __has_builtin(__builtin_amdgcn_global_load_async_to_lds_b128)
#define HAVE_ASYNC_LDS_BUILTIN 1
#endif
#if defined(__AMDGCN__) && __has_builtin(__builtin_amdgcn_s_wait_asynccnt)
#define HAVE_WAIT_ASYNC_BUILTIN 1
#endif

__device__ __forceinline__ void async_copy16(void* lds_dst, const void* gsrc) {
#if defined(HAVE_ASYNC_LDS_BUILTIN)
  __builtin_amdgcn_global_load_async_to_lds_b128(
      (gbl_v4i_ptr)gsrc, (lds_v4i_ptr)lds_dst, /*offset=*/0, /*cpol=*/0);
#elif defined(__AMDGCN__)
  unsigned loff = (unsigned)(unsigned long long)
      (__attribute__((address_space(3))) char*)lds_dst;
  unsigned long long ga = (unsigned long long)(uintptr_t)gsrc;
  asm volatile("global_load_async_to_lds_b128 %0, %1, off"
               :: "v"(loff), "v"(ga) : "memory");
#else
  *(float4*)lds_dst = *(const float4*)gsrc;   // host parse only
#endif
}

__device__ __forceinline__ void wait_async0() {
#if defined(HAVE_WAIT_ASYNC_BUILTIN)
  __builtin_amdgcn_s_wait_asynccnt(0);
#elif defined(__AMDGCN__)
  asm volatile("s_wait_asynccnt 0" ::: "memory");
#endif
}

// ---------------------------------------------------------------------------
// Device-scope sense-reversing barrier across the WGS_PER_BATCH workgroups of
// one batch. cnt/gen live in d_ws (zeroed by init kernel each launch).
// ---------------------------------------------------------------------------
__device__ __forceinline__ void batch_barrier(int* cnt, int* gen) {
  __syncthreads();
  if (threadIdx.x == 0) {
    __threadfence();  // release partial-result stores
    int g = __hip_atomic_load(gen, __ATOMIC_RELAXED, __HIP_MEMORY_SCOPE_AGENT);
    int prev = __hip_atomic_fetch_add(cnt, 1, __ATOMIC_ACQ_REL, __HIP_MEMORY_SCOPE_AGENT);
    if (prev == WGS_PER_BATCH - 1) {
      __hip_atomic_store(cnt, 0, __ATOMIC_RELAXED, __HIP_MEMORY_SCOPE_AGENT);
      __hip_atomic_fetch_add(gen, 1, __ATOMIC_ACQ_REL, __HIP_MEMORY_SCOPE_AGENT);
    } else {
      while (__hip_atomic_load(gen, __ATOMIC_ACQUIRE, __HIP_MEMORY_SCOPE_AGENT) == g) {
#if defined(__AMDGCN__)
        __builtin_amdgcn_s_sleep(8);
#endif
      }
    }
    __threadfence();  // acquire side before reading other WGs' partials
  }
  __syncthreads();
}

// ---------------------------------------------------------------------------
// Kernel: persistent cooperative FPS. Defined FIRST so the disassembly
// snippet shows its prologue + async-to-LDS staging loop. 4 WGs per batch;
// each stages its 25000-point slice (300 KB) into LDS once via async-to-LDS,
// keeps per-point running-min distances in 25 VGPRs/thread, and iterates
// 1024 argmax steps entirely out of LDS. Output rows written inline.
// ---------------------------------------------------------------------------
__global__ void __launch_bounds__(THREADS, 1)
fps_kernel(const float* __restrict__ xyz, const float* __restrict__ cm,
           const int* __restrict__ initf, const float* __restrict__ s_obj,
           float* __restrict__ out, float* __restrict__ partials,
           int* __restrict__ bar_cnt, int* __restrict__ bar_gen) {
  extern __shared__ float smem[];             // AoS xyz: NPPW * 3 floats
  __shared__ float rV[32];
  __shared__ int   rI[32];
  __shared__ float bc[4];                     // broadcast: fx, fy, fz, idx-bits

  const int tid       = threadIdx.x;
  const int b         = blockIdx.x / WGS_PER_BATCH;
  const int w         = blockIdx.x % WGS_PER_BATCH;
  const int sliceBase = w * NPPW;
  const float* gslice = xyz + ((size_t)b * NPTS + sliceBase) * 3;

  // ---- stage slice into LDS: contiguous 300000 B == 18750 b128 transfers
  {
    const int n16 = (NPPW * 3) / 4;           // float4 count (exact)
    for (int k = tid; k < n16; k += THREADS)
      async_copy16((float4*)smem + k, (const float4*)gslice + k);
    wait_async0();
  }
  __syncthreads();

  // ---- per-thread running-min distances live in VGPRs
  float dist[NPT];
#pragma unroll
  for (int k = 0; k < NPT; ++k) dist[k] = BIGF;

  int fidx = initf[b];                        // current farthest (batch-local)
  const float* fp0 = xyz + ((size_t)b * NPTS + fidx) * 3;
  float fx = fp0[0], fy = fp0[1], fz = fp0[2];

  const bool writer = (w == 0) && (tid == 0);
  float inv_s = 0.f;
  if (writer) inv_s = 1.0f / s_obj[b];

  int* cnt = bar_cnt + b;
  int* gen = bar_gen + b;

  for (int i = 0; i < NPOINT; ++i) {
    if (writer) {
      float cmv = cm[(size_t)b * NPTS + fidx];
      float4 row = make_float4(cmv, fx * inv_s, fy * inv_s, fz * inv_s);
      *(float4*)(out + (((size_t)b * NPOINT + i) << 2)) = row;
    }
    if (i == NPOINT - 1) break;               // uniform: no barrier imbalance

    // -- distance update + thread-local argmax (first-index tie break)
    float best = -1.0f;
    int   bestP = 0;
#pragma unroll
    for (int k = 0; k < NPT; ++k) {
      int p = tid + k * THREADS;
      if (p < NPPW) {
        float x = smem[3 * p + 0];
        float y = smem[3 * p + 1];
        float z = smem[3 * p + 2];
        float dx = x - fx, dy = y - fy, dz = z - fz;
        float d = fmaf(dz, dz, fmaf(dy, dy, dx * dx));
        d = fminf(dist[k], d);
        dist[k] = d;
        if (d > best) { best = d; bestP = p; }   // '>' keeps lowest index
      }
    }

    // -- wave32 shuffle reduction
    for (int o = 16; o > 0; o >>= 1) {
      float ov = __shfl_down(best, o);
      int   op = __shfl_down(bestP, o);
      if (ov > best || (ov == best && op < bestP)) { best = ov; bestP = op; }
    }
    if ((tid & 31) == 0) { rV[tid >> 5] = best; rI[tid >> 5] = bestP; }
    __syncthreads();
    if (tid < 32) {
      best = rV[tid]; bestP = rI[tid];
      for (int o = 16; o > 0; o >>= 1) {
        float ov = __shfl_down(best, o);
        int   op = __shfl_down(bestP, o);
        if (ov > best || (ov == best && op < bestP)) { best = ov; bestP = op; }
      }
      if (tid == 0) {
        // parity-double-buffered partial slot: {val, idx, x, y, z}
        float* slot = partials +
            ((((size_t)(i & 1) * BATCHES + b) * WGS_PER_BATCH + w) * 8);
        slot[0] = best;
        slot[1] = __int_as_float(sliceBase + bestP);
        slot[2] = smem[3 * bestP + 0];
        slot[3] = smem[3 * bestP + 1];
        slot[4] = smem[3 * bestP + 2];
      }
    }

    batch_barrier(cnt, gen);

    if (tid == 0) {
      float bv = -2.0f; int bi = 0; float bx = 0.f, by = 0.f, bz = 0.f;
      const float* pbase = partials +
          ((((size_t)(i & 1) * BATCHES + b) * WGS_PER_BATCH) * 8);
      for (int q = 0; q < WGS_PER_BATCH; ++q) {
        const float* s = pbase + q * 8;
        float v = s[0];
        int  ix = __float_as_int(s[1]);
        if (v > bv || (v == bv && ix < bi)) {
          bv = v; bi = ix; bx = s[2]; by = s[3]; bz = s[4];
        }
      }
      bc[0] = bx; bc[1] = by; bc[2] = bz; bc[3] = __int_as_float(bi);
    }
    __syncthreads();
    fx = bc[0]; fy = bc[1]; fz = bc[2]; fidx = __float_as_int(bc[3]);
  }
}

// ---------------------------------------------------------------------------
// Per-batch s_obj = max_i ||xyz_i - mean(xyz)||  (one block/batch)
// ---------------------------------------------------------------------------
__global__ void __launch_bounds__(THREADS, 1)
fps_stats(const float* __restrict__ xyz, float* __restrict__ s_obj) {
  __shared__ float sx[32], sy[32], sz[32], mx[32];
  __shared__ float cent[3];
  const int tid = threadIdx.x;
  const int b   = blockIdx.x;
  const float* base = xyz + (size_t)b * NPTS * 3;

  float ax = 0.f, ay = 0.f, az = 0.f;
  for (int p = tid; p < NPTS; p += THREADS) {
    ax += base[3 * p + 0];
    ay += base[3 * p + 1];
    az += base[3 * p + 2];
  }
  for (int o = 16; o > 0; o >>= 1) {
    ax += __shfl_down(ax, o);
    ay += __shfl_down(ay, o);
    az += __shfl_down(az, o);
  }
  if ((tid & 31) == 0) { sx[tid >> 5] = ax; sy[tid >> 5] = ay; sz[tid >> 5] = az; }
  __syncthreads();
  if (tid < 32) {
    ax = sx[tid]; ay = sy[tid]; az = sz[tid];
    for (int o = 16; o > 0; o >>= 1) {
      ax += __shfl_down(ax, o);
      ay += __shfl_down(ay, o);
      az += __shfl_down(az, o);
    }
    if (tid == 0) {
      cent[0] = ax / (float)NPTS;
      cent[1] = ay / (float)NPTS;
      cent[2] = az / (float)NPTS;
    }
  }
  __syncthreads();
  const float cx = cent[0], cy = cent[1], cz = cent[2];

  float m = 0.f;
  for (int p = tid; p < NPTS; p += THREADS) {
    float dx = base[3 * p + 0] - cx;
    float dy = base[3 * p + 1] - cy;
    float dz = base[3 * p + 2] - cz;
    float d2 = fmaf(dz, dz, fmaf(dy, dy, dx * dx));
    m = fmaxf(m, d2);
  }
  for (int o = 16; o > 0; o >>= 1) m = fmaxf(m, __shfl_down(m, o));
  if ((tid & 31) == 0) mx[tid >> 5] = m;
  __syncthreads();
  if (tid < 32) {
    m = mx[tid];
    for (int o = 16; o > 0; o >>= 1) m = fmaxf(m, __shfl_down(m, o));
    if (tid == 0) s_obj[b] = sqrtf(m);
  }
}

// ---------------------------------------------------------------------------
// Zero the inter-WG barrier state in workspace (ws is poisoned each run).
// ---------------------------------------------------------------------------
__global__ void fps_init_ws(int* cnt, int* gen) {
  int t = threadIdx.x;
  if (t < BATCHES) { cnt[t] = 0; gen[t] = 0; }
}

// ---------------------------------------------------------------------------
// Host entry
// ---------------------------------------------------------------------------
extern "C" void kernel_launch(void* const* d_in, const int* in_sizes, int n_in,
                              void* d_out, int out_size, void* d_ws, size_t ws_size,
                              hipStream_t stream) {
  (void)in_sizes; (void)n_in; (void)out_size; (void)ws_size;
  const float* mesh  = (const float*)d_in[0];   // [16,100000,3] f32
  const float* cmap  = (const float*)d_in[1];   // [16,100000,1] f32
  const int*   initf = (const int*)d_in[2];     // [16] i32
  // d_in[3] = npoint (==1024), baked in as NPOINT.
  float* out = (float*)d_out;                   // [16,1024,4] f32

  // Allow >64KB dynamic LDS for the FPS kernel (idempotent; capture-safe —
  // it is a function attribute update, not a stream operation).
  (void)hipFuncSetAttribute((const void*)fps_kernel,
                            hipFuncAttributeMaxDynamicSharedMemorySize,
                            (int)SLICE_BYTES);

  // workspace layout (floats): [s_obj:16][partials: 2*16*4*8][cnt:16 gen:16 ints]
  float* wsf      = (float*)d_ws;
  float* s_obj    = wsf;
  float* partials = wsf + BATCHES;
  int*   ibase    = (int*)(wsf + BATCHES + 2 * BATCHES * WGS_PER_BATCH * 8);
  int*   cnt      = ibase;
  int*   gen      = ibase + BATCHES;

  hipLaunchKernelGGL(fps_init_ws, dim3(1), dim3(32), 0, stream, cnt, gen);
  hipLaunchKernelGGL(fps_stats, dim3(BATCHES), dim3(THREADS), 0, stream, mesh, s_obj);
  hipLaunchKernelGGL(fps_kernel, dim3(BATCHES * WGS_PER_BATCH), dim3(THREADS),
                     SLICE_BYTES, stream,
                     mesh, cmap, initf, s_obj, out, partials, cnt, gen);
}